// Model_Both_46222438039983
// MI455X (gfx1250) — compile-verified
//
#include <hip/hip_runtime.h>

#define T_DIM   12
#define WINDOW  5
#define N_NODES 4000
#define E_EDGES 32000
#define K_HEADS 8
#define HD      8
#define D_MODEL 64
#define B_BATCH 4
#define LN_EPS  1e-12f

typedef float v8f __attribute__((ext_vector_type(8)));
typedef float v2f __attribute__((ext_vector_type(2)));
typedef unsigned int v4u __attribute__((ext_vector_type(4)));
typedef int v4i __attribute__((ext_vector_type(4)));
typedef int v8i __attribute__((ext_vector_type(8)));

#if __has_builtin(__builtin_amdgcn_tensor_load_to_lds) && __has_builtin(__builtin_amdgcn_s_wait_tensorcnt)
#define USE_TDM 1
#else
#define USE_TDM 0
#endif

// ------------------------------------------------------------------
// Stage 1: q/k/v = concat(X, STE) @ W{q,k,v} + b   (q scaled 1/sqrt(HD))
// Block = 384 threads = 12 waves; wave w owns (matrix m = w>>2, coltile
// c = w&3) and keeps its 64 B-fragment registers live across all node
// tiles. A tile (16 x 128 f32) staged in LDS with row stride 132 floats
// (132 mod 64 = 4 -> conflict-free ds_load_b64 fragment reads).
// A-tile staging uses the Tensor Data Mover: two 2D tile descriptors
// (X half -> LDS col 0, STE half -> LDS col 64) with LDS row padding of
// 68 DWORDs every 64 DWORDs to realize the 132-float stride.
// ------------------------------------------------------------------
#define S1_TILES_PER_BLOCK 10

__global__ __launch_bounds__(384)
void qkv_kernel(const float* __restrict__ X, const float* __restrict__ STE,
                const float* __restrict__ Wq, const float* __restrict__ bq,
                const float* __restrict__ Wk, const float* __restrict__ bk,
                const float* __restrict__ Wv, const float* __restrict__ bv,
                float* __restrict__ qo, float* __restrict__ ko, float* __restrict__ vo)
{
    __shared__ float As[16 * 132];

    const int blocksPerSlice = (N_NODES / 16) / S1_TILES_PER_BLOCK;   // 25
    const int slice = blockIdx.x / blocksPerSlice;                    // b*T + t
    const int tile0 = (blockIdx.x % blocksPerSlice) * S1_TILES_PER_BLOCK;

    const int tid  = threadIdx.x;
    const int lane = tid & 31;
    const int wave = tid >> 5;      // 0..11
    const int half = lane >> 4;     // K half selector
    const int l16  = lane & 15;

    const size_t sliceBase = (size_t)slice * N_NODES * D_MODEL;

    const int m = wave >> 2;        // 0=q 1=k 2=v
    const int c = wave & 3;         // 16-col tile
    const float* W   = (m == 0) ? Wq : ((m == 1) ? Wk : Wv);
    const float* bbp = (m == 0) ? bq : ((m == 1) ? bk : bv);
    float* op = ((m == 0) ? qo : ((m == 1) ? ko : vo)) + sliceBase;

    // B fragments once per block: lane holds W[k][c*16+l16], k = 4s+2*half+{0,1}
    float bf[64];
    #pragma unroll
    for (int s = 0; s < 32; ++s) {
        int kb = 4 * s + 2 * half;
        bf[2 * s]     = W[(size_t)(kb)     * 64 + c * 16 + l16];
        bf[2 * s + 1] = W[(size_t)(kb + 1) * 64 + c * 16 + l16];
    }
    const float bias  = bbp[c * 16 + l16];
    const float scale = (m == 0) ? 0.35355339059327373f : 1.0f;  // 1/sqrt(HD) folded into q

#if USE_TDM
    // D# group1 (invariant): data_size=4B (2), pad_enable, pad_interval=64
    // DWORDs (enc 5), pad_amount=68 DWORDs (enc 67); tensor_dim0=64,
    // tensor_dim1=16, tile 64x16, dim0 stride=64 elements.
    v8i g1;
    g1[0] = (int)((2u << 16) | (1u << 20) | (5u << 22) | (67u << 25));
    g1[1] = (int)(64u << 16);   // tensor_dim0[15:0] at [31:16]
    g1[2] = (int)(16u << 16);   // tensor_dim0[31:16]=0, tensor_dim1[15:0]=16
    g1[3] = (int)(64u << 16);   // tensor_dim1[31:16]=0, tile_dim0=64
    g1[4] = 16;                 // tile_dim1=16, tile_dim2=0
    g1[5] = 64;                 // tensor_dim0_stride low32 = 64
    g1[6] = 0;
    g1[7] = 0;
    const v4i gz  = {};
    const v8i gz8 = {};
#endif

    for (int nt = 0; nt < S1_TILES_PER_BLOCK; ++nt) {
        const int n0 = (tile0 + nt) * 16;
        __syncthreads();   // previous tile fully consumed before overwrite
#if USE_TDM
        if (wave == 0) {
            unsigned long long gaX =
                (unsigned long long)(const void*)(X + sliceBase + (size_t)n0 * D_MODEL);
            unsigned long long gaS =
                (unsigned long long)(const void*)(STE + sliceBase + (size_t)n0 * D_MODEL);
            v4u g0x, g0s;
            g0x[0] = 1u;                       // count=1 valid descriptor
            g0x[1] = 0u;                       // lds_addr (As at LDS offset 0)
            g0x[2] = (unsigned)(gaX & 0xFFFFFFFFu);
            g0x[3] = (unsigned)(gaX >> 32) | (2u << 30);   // type=2 (image)
            g0s[0] = 1u;
            g0s[1] = 64u * 4u;                 // STE half at LDS col 64
            g0s[2] = (unsigned)(gaS & 0xFFFFFFFFu);
            g0s[3] = (unsigned)(gaS >> 32) | (2u << 30);
            __builtin_amdgcn_tensor_load_to_lds(g0x, g1, gz, gz, gz8, 0);
            __builtin_amdgcn_tensor_load_to_lds(g0s, g1, gz, gz, gz8, 0);
            __builtin_amdgcn_s_wait_tensorcnt(0);
        }
#else
        if (tid < 256) {
            int f   = tid * 8;
            int row = f >> 7;
            int col = f & 127;
            const float* sp = (col < 64)
                ? (X   + sliceBase + (size_t)(n0 + row) * D_MODEL + col)
                : (STE + sliceBase + (size_t)(n0 + row) * D_MODEL + (col - 64));
            float4 p0 = *(const float4*)(sp);
            float4 p1 = *(const float4*)(sp + 4);
            float* dp = &As[row * 132 + col];
            *(float4*)(dp)     = p0;
            *(float4*)(dp + 4) = p1;
        }
#endif
        __syncthreads();

#if USE_TDM
        // The TDM wrote As behind the compiler's back: escape the LDS
        // pointer and clobber memory so the ds_load fragment reads below
        // are not folded away as loads of uninitialized memory.
        {
            const float* esc = &As[0];
            asm volatile("" : "+v"(esc) : : "memory");
        }
#endif

        v8f acc = {};
        #pragma unroll
        for (int s = 0; s < 32; ++s) {
            const float* ap = &As[l16 * 132 + 4 * s + 2 * half];
            v2f a;   a.x = ap[0];        a.y = ap[1];
            v2f bv2; bv2.x = bf[2 * s];  bv2.y = bf[2 * s + 1];
            acc = __builtin_amdgcn_wmma_f32_16x16x4_f32(
                      false, a, false, bv2, (short)0, acc, false, false);
        }

        #pragma unroll
        for (int i = 0; i < 8; ++i) {
            int row = n0 + i + 8 * half;
            op[(size_t)row * D_MODEL + c * 16 + l16] = (acc[i] + bias) * scale;
        }
    }
}

// ------------------------------------------------------------------
// Stage 2: windowed edge attention with segment softmax over sorted dst.
// One block = (node n, batch b), 96 threads = t(12) x head(8).
// Three passes per shift (max / denom / weighted V), dots recomputed.
// ------------------------------------------------------------------
__global__ __launch_bounds__(96)
void attn_kernel(const float* __restrict__ q, const float* __restrict__ kmat,
                 const float* __restrict__ vmat, const int* __restrict__ src,
                 const int* __restrict__ dst, float* __restrict__ res)
{
    const int n   = blockIdx.x;
    const int b   = blockIdx.y;
    const int tid = threadIdx.x;
    const int kh  = tid & 7;
    const int t   = tid >> 3;

    // edge range [e0, e1) for destination n (dst is sorted)
    int lo = 0, hi = E_EDGES;
    while (lo < hi) { int mid = (lo + hi) >> 1; if (dst[mid] < n) lo = mid + 1; else hi = mid; }
    const int e0 = lo;
    hi = E_EDGES;
    while (lo < hi) { int mid = (lo + hi) >> 1; if (dst[mid] < n + 1) lo = mid + 1; else hi = mid; }
    const int e1 = lo;

    const size_t bBase = (size_t)b * T_DIM * N_NODES * D_MODEL;

    float q8[8];
    {
        const float* qp = q + bBase + ((size_t)t * N_NODES + n) * D_MODEL + kh * 8;
        #pragma unroll
        for (int i = 0; i < 8; ++i) q8[i] = qp[i];
    }

    float r8[8] = {};

    for (int s = 0; s < WINDOW; ++s) {
        int ts = t + s - (WINDOW - 1) / 2;
        if (ts < 0)         ts += WINDOW;
        if (ts > T_DIM - 1) ts -= WINDOW;
        const float* kb = kmat + bBase + (size_t)ts * N_NODES * D_MODEL + kh * 8;
        const float* vb = vmat + bBase + (size_t)ts * N_NODES * D_MODEL + kh * 8;

        float mx = -3.402823466e+38f;
        for (int e = e0; e < e1; ++e) {
            const float* kp = kb + (size_t)src[e] * D_MODEL;
            float d = 0.f;
            #pragma unroll
            for (int i = 0; i < 8; ++i) d += kp[i] * q8[i];
            mx = fmaxf(mx, d);
        }
        float denom = 0.f;
        for (int e = e0; e < e1; ++e) {
            const float* kp = kb + (size_t)src[e] * D_MODEL;
            float d = 0.f;
            #pragma unroll
            for (int i = 0; i < 8; ++i) d += kp[i] * q8[i];
            denom += __expf(d - mx);
        }
        const float inv = (denom > 0.f) ? (1.0f / denom) : 0.f;
        for (int e = e0; e < e1; ++e) {
            const int sn = src[e];
            const float* kp = kb + (size_t)sn * D_MODEL;
            float d = 0.f;
            #pragma unroll
            for (int i = 0; i < 8; ++i) d += kp[i] * q8[i];
            const float w = __expf(d - mx) * inv;
            const float* vp = vb + (size_t)sn * D_MODEL;
            #pragma unroll
            for (int i = 0; i < 8; ++i) r8[i] += w * vp[i];
        }
    }

    float* rp = res + bBase + ((size_t)t * N_NODES + n) * D_MODEL + kh * 8;
    #pragma unroll
    for (int i = 0; i < 8; ++i) rp[i] = r8[i] * (1.0f / WINDOW);
}

// ------------------------------------------------------------------
// Stage 3: hidden = res @ Wo + bo ; out = LayerNorm(hidden + X)
// One block = 32 rows. 8 waves = 2 row-tiles x 4 col-tiles (WMMA f32).
// Hidden goes through LDS, then 32 threads do residual + LN per row.
// ------------------------------------------------------------------
__global__ __launch_bounds__(256)
void out_ln_kernel(const float* __restrict__ res, const float* __restrict__ Wo,
                   const float* __restrict__ bo, const float* __restrict__ X,
                   const float* __restrict__ ln_scale, const float* __restrict__ ln_bias,
                   float* __restrict__ out)
{
    __shared__ float As[32 * 68];
    __shared__ float Hs[32 * 66];

    const int tid  = threadIdx.x;
    const int lane = tid & 31;
    const int wave = tid >> 5;
    const int half = lane >> 4;
    const int l16  = lane & 15;
    const size_t row0 = (size_t)blockIdx.x * 32;

    {
        int f   = tid * 8;
        int row = f >> 6;
        int col = f & 63;
        const float* sp = res + (row0 + row) * D_MODEL + col;
        float4 p0 = *(const float4*)sp;
        float4 p1 = *(const float4*)(sp + 4);
        float* dp = &As[row * 68 + col];
        *(float4*)(dp)     = p0;
        *(float4*)(dp + 4) = p1;
    }
    __syncthreads();

    const int rt = wave >> 2;
    const int c  = wave & 3;

    float bf[32];
    #pragma unroll
    for (int s = 0; s < 16; ++s) {
        int kb = 4 * s + 2 * half;
        bf[2 * s]     = Wo[(size_t)(kb)     * 64 + c * 16 + l16];
        bf[2 * s + 1] = Wo[(size_t)(kb + 1) * 64 + c * 16 + l16];
    }

    v8f acc = {};
    #pragma unroll
    for (int s = 0; s < 16; ++s) {
        const float* ap = &As[(rt * 16 + l16) * 68 + 4 * s + 2 * half];
        v2f a;   a.x = ap[0];        a.y = ap[1];
        v2f bv2; bv2.x = bf[2 * s];  bv2.y = bf[2 * s + 1];
        acc = __builtin_amdgcn_wmma_f32_16x16x4_f32(
                  false, a, false, bv2, (short)0, acc, false, false);
    }

    const float bias = bo[c * 16 + l16];
    #pragma unroll
    for (int i = 0; i < 8; ++i)
        Hs[(rt * 16 + i + 8 * half) * 66 + c * 16 + l16] = acc[i] + bias;
    __syncthreads();

    if (tid < 32) {
        const size_t g = (row0 + tid) * D_MODEL;
        float hrow[64];
        float sum = 0.f;
        #pragma unroll
        for (int j = 0; j < 64; ++j) {
            float h = Hs[tid * 66 + j] + X[g + j];
            hrow[j] = h;
            sum += h;
        }
        const float mu = sum * (1.0f / 64.0f);
        float sumsq = 0.f;
        #pragma unroll
        for (int j = 0; j < 64; ++j) {
            float d = hrow[j] - mu;
            sumsq += d * d;
        }
        const float rstd = rsqrtf(sumsq * (1.0f / 64.0f) + LN_EPS);
        #pragma unroll
        for (int j = 0; j < 64; ++j)
            out[g + j] = (hrow[j] - mu) * rstd * ln_scale[j] + ln_bias[j];
    }
}

// ------------------------------------------------------------------
extern "C" void kernel_launch(void* const* d_in, const int* in_sizes, int n_in,
                              void* d_out, int out_size, void* d_ws, size_t ws_size,
                              hipStream_t stream)
{
    const float* X        = (const float*)d_in[0];
    const float* STE      = (const float*)d_in[1];
    const float* Wq       = (const float*)d_in[2];
    const float* bq       = (const float*)d_in[3];
    const float* Wk       = (const float*)d_in[4];
    const float* bk       = (const float*)d_in[5];
    const float* Wv       = (const float*)d_in[6];
    const float* bv       = (const float*)d_in[7];
    const float* Wo       = (const float*)d_in[8];
    const float* bo       = (const float*)d_in[9];
    const float* ln_scale = (const float*)d_in[10];
    const float* ln_bias  = (const float*)d_in[11];
    const int*   src      = (const int*)d_in[12];
    const int*   dst      = (const int*)d_in[13];
    float* out = (float*)d_out;

    const size_t S = (size_t)B_BATCH * T_DIM * N_NODES * D_MODEL; // 12.288M floats
    float* q   = (float*)d_ws;
    float* k   = q + S;
    float* v   = k + S;
    float* res = v + S;

    const int blocksPerSlice = (N_NODES / 16) / S1_TILES_PER_BLOCK; // 25
    dim3 g1(blocksPerSlice * B_BATCH * T_DIM);                      // 1200
    qkv_kernel<<<g1, 384, 0, stream>>>(X, STE, Wq, bq, Wk, bk, Wv, bv, q, k, v);

    dim3 g2(N_NODES, B_BATCH);                                      // 16000
    attn_kernel<<<g2, 96, 0, stream>>>(q, k, v, src, dst, res);

    dim3 g3((unsigned)((S / D_MODEL) / 32));                        // 6000
    out_ln_kernel<<<g3, 256, 0, stream>>>(res, Wo, bo, X, ln_scale, ln_bias, out);
}